// DistillationLoss_52982716564146
// MI455X (gfx1250) — compile-verified
//
#include <hip/hip_runtime.h>
#include <math.h>

#define TILE 1024           // floats per tile per tensor (4 KB)
#define BLK  256            // 8 waves (wave32)
#define NW   (BLK / 32)

// ---- CDNA5 async global->LDS path (probe via __has_builtin) ----------------
#if defined(__has_builtin)
#  if __has_builtin(__builtin_amdgcn_global_load_async_to_lds_b128)
#    define HAVE_ASYNC 1
#  endif
#endif

typedef int v4i __attribute__((vector_size(16)));
typedef __attribute__((address_space(1))) v4i* gas_v4i_ptr;   // global
typedef __attribute__((address_space(3))) v4i* las_v4i_ptr;   // LDS

#if defined(HAVE_ASYNC)
#  if __has_builtin(__builtin_amdgcn_s_wait_asynccnt)
#    define WAIT_ASYNC(n) do { __builtin_amdgcn_s_wait_asynccnt(n); asm volatile("" ::: "memory"); } while (0)
#  else
#    define WAIT_ASYNC(n) asm volatile("s_wait_asynccnt %0" :: "i"(n) : "memory")
#  endif
#endif

__device__ __forceinline__ void async_f4(const float* g, float* l) {
#if defined(HAVE_ASYNC)
  __builtin_amdgcn_global_load_async_to_lds_b128(
      (gas_v4i_ptr)(v4i*)(uintptr_t)g,
      (las_v4i_ptr)(v4i*)l,
      0, 0);
#else
  (void)g; (void)l;
#endif
}

// ---- helpers ---------------------------------------------------------------
__device__ __forceinline__ float pow4f(float x) { x *= x; return x * x; }

__device__ __forceinline__ float waveSum(float v) {
#pragma unroll
  for (int o = 16; o > 0; o >>= 1) v += __shfl_xor(v, o, 32);
  return v;
}
__device__ __forceinline__ float waveMax(float v) {
#pragma unroll
  for (int o = 16; o > 0; o >>= 1) v = fmaxf(v, __shfl_xor(v, o, 32));
  return v;
}
__device__ __forceinline__ float blockSum(float v, float* sred) {
  v = waveSum(v);
  __syncthreads();                       // protect scratch from previous use
  if ((threadIdx.x & 31) == 0) sred[threadIdx.x >> 5] = v;
  __syncthreads();
  float r = sred[0];
#pragma unroll
  for (int i = 1; i < NW; ++i) r += sred[i];
  return r;                              // broadcast to all threads, fixed order
}
__device__ __forceinline__ float blockMax(float v, float* sred) {
  v = waveMax(v);
  __syncthreads();
  if ((threadIdx.x & 31) == 0) sred[threadIdx.x >> 5] = v;
  __syncthreads();
  float r = sred[0];
#pragma unroll
  for (int i = 1; i < NW; ++i) r = fmaxf(r, sred[i]);
  return r;
}

// Online-softmax chunk update (T = 4 hardcoded: 1/T = 0.25, e^x = (e^{x/4})^4)
__device__ __forceinline__ void upd(const float4 sv, const float4 tv,
                                    float& ms, float& ssT, float& ss1,
                                    float& mt, float& stT, float& aa) {
  // student: needs Σ e^{(s-m)/4} and Σ e^{s-m}
  float cm = fmaxf(fmaxf(sv.x, sv.y), fmaxf(sv.z, sv.w));
  float nm = fmaxf(ms, cm);
  float fT = __expf((ms - nm) * 0.25f);          // -inf -> 0 on first chunk
  float e0 = __expf((sv.x - nm) * 0.25f);
  float e1 = __expf((sv.y - nm) * 0.25f);
  float e2 = __expf((sv.z - nm) * 0.25f);
  float e3 = __expf((sv.w - nm) * 0.25f);
  ssT = fmaf(ssT, fT, (e0 + e1) + (e2 + e3));
  ss1 = fmaf(ss1, pow4f(fT), (pow4f(e0) + pow4f(e1)) + (pow4f(e2) + pow4f(e3)));
  ms = nm;
  // teacher: needs Σ e^{(t-m)/4} and Σ e^{(t-m)/4} (t - s)
  float cmt = fmaxf(fmaxf(tv.x, tv.y), fmaxf(tv.z, tv.w));
  float nmt = fmaxf(mt, cmt);
  float gT = __expf((mt - nmt) * 0.25f);
  float q0 = __expf((tv.x - nmt) * 0.25f);
  float q1 = __expf((tv.y - nmt) * 0.25f);
  float q2 = __expf((tv.z - nmt) * 0.25f);
  float q3 = __expf((tv.w - nmt) * 0.25f);
  stT = fmaf(stT, gT, (q0 + q1) + (q2 + q3));
  aa  = fmaf(aa,  gT, (q0 * (tv.x - sv.x) + q1 * (tv.y - sv.y)) +
                      (q2 * (tv.z - sv.z) + q3 * (tv.w - sv.w)));
  mt = nmt;
}

// ---- kernel 1: one block per row, fused online softmax + KL + NLL ----------
__global__ __launch_bounds__(BLK)
void distill_row_kernel(const float* __restrict__ S, const float* __restrict__ T,
                        const int* __restrict__ labels,
                        float* __restrict__ ws, int V) {
  __shared__ float lds_s[2][TILE];
  __shared__ float lds_t[2][TILE];
  __shared__ float sred[NW];

  const int n = blockIdx.x;
  const int tid = threadIdx.x;
  const float* __restrict__ srow = S + (size_t)n * (size_t)V;
  const float* __restrict__ trow = T + (size_t)n * (size_t)V;

  float ms = -__builtin_inff(), ssT = 0.f, ss1 = 0.f;
  float mt = -__builtin_inff(), stT = 0.f, aa  = 0.f;

  const int NT = (V + TILE - 1) / TILE;

#if defined(HAVE_ASYNC)
  {  // prologue: tile 0 -> buffer 0 (always issue; clamp keeps addr in range)
    int idx = tid * 4;
    int c = idx < V ? idx : V - 4;
    async_f4(srow + c, &lds_s[0][tid * 4]);
    async_f4(trow + c, &lds_t[0][tid * 4]);
  }
  for (int tt = 0; tt < NT; ++tt) {
    const int cb = tt & 1;
    if (tt + 1 < NT) {
      int idx = (tt + 1) * TILE + tid * 4;
      int c = idx < V ? idx : V - 4;
      async_f4(srow + c, &lds_s[cb ^ 1][tid * 4]);
      async_f4(trow + c, &lds_t[cb ^ 1][tid * 4]);
      WAIT_ASYNC(2);                    // tile tt done; tile tt+1 in flight
    } else {
      WAIT_ASYNC(0);
    }
    int idx = tt * TILE + tid * 4;
    if (idx < V) {
      float4 sv = *(const float4*)&lds_s[cb][tid * 4];
      float4 tv = *(const float4*)&lds_t[cb][tid * 4];
      upd(sv, tv, ms, ssT, ss1, mt, stT, aa);
    }
    asm volatile("" ::: "memory");      // keep issue/compute pipeline ordered
  }
#else
  for (int idx = tid * 4; idx < V; idx += BLK * 4) {
    int pf = idx + BLK * 4 * 8;
    if (pf < V) { __builtin_prefetch(srow + pf, 0, 0); __builtin_prefetch(trow + pf, 0, 0); }
    float4 sv = *(const float4*)(srow + idx);
    float4 tv = *(const float4*)(trow + idx);
    upd(sv, tv, ms, ssT, ss1, mt, stT, aa);
  }
#endif

  // merge lane-local stats: rescale to block max, then linear sums
  float Ms = blockMax(ms, sred);
  float Mt = blockMax(mt, sred);
  float fs = __expf((ms - Ms) * 0.25f);   // 0 if lane saw nothing (ms = -inf)
  float ft = __expf((mt - Mt) * 0.25f);
  float SsT = blockSum(ssT * fs, sred);
  float Ss1 = blockSum(ss1 * pow4f(fs), sred);
  float StT = blockSum(stT * ft, sred);
  float A   = blockSum(aa  * ft, sred);

  if (tid == 0) {
    // KL_row = (1/T)*Σp_t(t-s) - (Mt-Ms)/T - log StT + log SsT
    float kl_row = 0.25f * (A / StT) - 0.25f * (Mt - Ms) - __logf(StT) + __logf(SsT);
    int lab = labels[n];
    float nll = (Ms + __logf(Ss1)) - srow[lab];   // -log_softmax(s)[lab]
    float valid = (lab != 0) ? 1.0f : 0.0f;
    ws[3 * n + 0] = kl_row;
    ws[3 * n + 1] = valid * nll;
    ws[3 * n + 2] = valid;
  }
}

// ---- kernel 2: deterministic final reduction over N rows -------------------
__global__ __launch_bounds__(BLK)
void distill_final_kernel(const float* __restrict__ ws, float* __restrict__ out, int N) {
  __shared__ float sred[NW];
  float kl = 0.f, nll = 0.f, cnt = 0.f;
  for (int i = threadIdx.x; i < N; i += BLK) {
    kl  += ws[3 * i + 0];
    nll += ws[3 * i + 1];
    cnt += ws[3 * i + 2];
  }
  kl  = blockSum(kl, sred);
  nll = blockSum(nll, sred);
  cnt = blockSum(cnt, sred);
  if (threadIdx.x == 0) {
    const float T2 = 16.0f;            // TEMPERATURE^2
    float distill = (kl / (float)N) * T2;
    float task = nll / fmaxf(cnt, 1.0f);
    float total = 0.7f * distill + 0.3f * task;   // ALPHA = 0.7
    out[0] = total;
    out[1] = distill;
    out[2] = task;
  }
}

// ---- host launcher ---------------------------------------------------------
extern "C" void kernel_launch(void* const* d_in, const int* in_sizes, int n_in,
                              void* d_out, int out_size, void* d_ws, size_t ws_size,
                              hipStream_t stream) {
  const float* S = (const float*)d_in[0];
  const float* T = (const float*)d_in[1];
  const int*   L = (const int*)d_in[2];
  float* out = (float*)d_out;
  float* ws  = (float*)d_ws;
  const int N = in_sizes[2];             // B*S = 4096 rows
  const int V = in_sizes[0] / N;         // 32000
  distill_row_kernel<<<N, BLK, 0, stream>>>(S, T, L, ws, V);
  distill_final_kernel<<<1, BLK, 0, stream>>>(ws, out, N);
}